// myGNN2_17016660427425
// MI455X (gfx1250) — compile-verified
//
#include <hip/hip_runtime.h>
#include <math.h>

#define NNODES  50000
#define NEDGES  800000
#define C       64
#define NGRAPHS 64

typedef __attribute__((ext_vector_type(2))) float v2f;
typedef __attribute__((ext_vector_type(8))) float v8f;

// ---------------- zero helpers ----------------
__global__ void k_zero_f4(float4* __restrict__ p, int n4) {
    int i = blockIdx.x * blockDim.x + threadIdx.x;
    if (i < n4) p[i] = make_float4(0.f, 0.f, 0.f, 0.f);
}

__global__ void k_zero_i(int* __restrict__ p, int n) {
    int i = blockIdx.x * blockDim.x + threadIdx.x;
    if (i < n) p[i] = 0;
}

// ---------------- edge scatter-add: agg[dst] += h[src] ----------------
// 16 lanes per edge, float4 per lane -> 256B coalesced gather per edge.
// L2-resident destination (12.8MB), native fp32 atomics.
__global__ __launch_bounds__(256) void k_scatter(const float* __restrict__ h,
                                                 const int* __restrict__ src,
                                                 const int* __restrict__ dst,
                                                 float* __restrict__ agg) {
    int idx = blockIdx.x * blockDim.x + threadIdx.x;   // exactly NEDGES*16 threads
    int e   = idx >> 4;
    int c4  = (idx & 15) << 2;
    int s = src[e];
    int d = dst[e];
    const float4 v = *(const float4*)(h + (size_t)s * C + c4);
    float* base = agg + (size_t)d * C + c4;
    __hip_atomic_fetch_add(base + 0, v.x, __ATOMIC_RELAXED, __HIP_MEMORY_SCOPE_AGENT);
    __hip_atomic_fetch_add(base + 1, v.y, __ATOMIC_RELAXED, __HIP_MEMORY_SCOPE_AGENT);
    __hip_atomic_fetch_add(base + 2, v.z, __ATOMIC_RELAXED, __HIP_MEMORY_SCOPE_AGENT);
    __hip_atomic_fetch_add(base + 3, v.w, __ATOMIC_RELAXED, __HIP_MEMORY_SCOPE_AGENT);
}

// ---------------- GraphConv GEMM: H = relu(X@Wr + G@Wg + b) ----------------
// 128 threads = 4 waves; wave w owns output columns [16w, 16w+16).
// Full-precision f32 WMMA 16x16x4, K=64 as 16 chained steps per matrix.
// A layout (16x4 f32): lanes 0-15 hold K=0,1 in v[0],v[1]; lanes 16-31 K=2,3.
// B layout (4x16 f32): lanes 0-15 -> K=0(v0),K=1(v1); lanes 16-31 -> K=2,3.
// C/D: VGPR r = row r (lanes 0-15) / row r+8 (lanes 16-31), N = lane&15.
__global__ __launch_bounds__(128) void k_conv_gemm(const float* __restrict__ X,
                                                   const float* __restrict__ G,
                                                   const float* __restrict__ Wr,
                                                   const float* __restrict__ Wg,
                                                   const float* __restrict__ bias,
                                                   float* __restrict__ H,
                                                   int ntiles) {
    const int lane = threadIdx.x & 31;
    const int wave = threadIdx.x >> 5;      // n-tile within block
    const int m    = lane & 15;             // A row / C column index
    const int half = lane >> 4;             // 0 or 1 (K pair selector)
    const int ncol = wave * 16 + m;         // output column handled for B/C

    // Preload weight (B) fragments for both matrices: resident in VGPRs.
    v2f br[16], bg[16];
#pragma unroll
    for (int kk = 0; kk < 16; ++kk) {
        const int k0 = kk * 4 + half * 2;
        br[kk][0] = Wr[(k0 + 0) * C + ncol];
        br[kk][1] = Wr[(k0 + 1) * C + ncol];
        bg[kk][0] = Wg[(k0 + 0) * C + ncol];
        bg[kk][1] = Wg[(k0 + 1) * C + ncol];
    }
    const float bv = bias[ncol];

    for (int tile = blockIdx.x; tile < ntiles; tile += gridDim.x) {
        const float* xrow = X + ((size_t)tile * 16 + m) * C + half * 2;
        const float* grow = G + ((size_t)tile * 16 + m) * C + half * 2;
        v8f accr = {};
        v8f accg = {};
#pragma unroll
        for (int kk = 0; kk < 16; ++kk) {
            v2f a = *(const v2f*)(xrow + kk * 4);
            v2f g = *(const v2f*)(grow + kk * 4);
            accr = __builtin_amdgcn_wmma_f32_16x16x4_f32(
                false, a, false, br[kk], (short)0, accr, false, false);
            accg = __builtin_amdgcn_wmma_f32_16x16x4_f32(
                false, g, false, bg[kk], (short)0, accg, false, false);
        }
#pragma unroll
        for (int r = 0; r < 8; ++r) {
            const int mrow = r + half * 8;
            float v = accr[r] + accg[r] + bv;
            v = v > 0.f ? v : 0.f;
            H[((size_t)tile * 16 + mrow) * C + ncol] = v;
        }
    }
}

// ---------------- per-node MLP head + mid softmax + segment-max ----------------
__global__ __launch_bounds__(256) void k_post(const float* __restrict__ H2,
                                              const int* __restrict__ batch,
                                              const float* __restrict__ fc1w,
                                              const float* __restrict__ fc1b,
                                              const float* __restrict__ fc2w,
                                              const float* __restrict__ fc2b,
                                              float* __restrict__ out_mid,
                                              int* __restrict__ pooled_i) {
    __shared__ float s_fc1w[C * 8];
    __shared__ float s_fc1b[8];
    __shared__ float s_fc2w[16];
    __shared__ float s_fc2b[2];
    const int t = threadIdx.x;
    if (t < C * 8) s_fc1w[t] = fc1w[t];
    if (t < 8)     s_fc1b[t] = fc1b[t];
    if (t < 16)    s_fc2w[t] = fc2w[t];
    if (t < 2)     s_fc2b[t] = fc2b[t];
    __syncthreads();

    const int node = blockIdx.x * blockDim.x + t;
    if (node >= NNODES) return;

    float h[C];
    const float4* row = (const float4*)(H2 + (size_t)node * C);
#pragma unroll
    for (int i = 0; i < C / 4; ++i) {
        float4 v = row[i];
        h[4 * i + 0] = v.x; h[4 * i + 1] = v.y;
        h[4 * i + 2] = v.z; h[4 * i + 3] = v.w;
    }

    // mid = softmax(h[:8])
    float mx = h[0];
#pragma unroll
    for (int j = 1; j < 8; ++j) mx = fmaxf(mx, h[j]);
    float ex[8], s = 0.f;
#pragma unroll
    for (int j = 0; j < 8; ++j) { ex[j] = expf(h[j] - mx); s += ex[j]; }
    const float inv = 1.f / s;
#pragma unroll
    for (int j = 0; j < 8; ++j) out_mid[(size_t)node * 8 + j] = ex[j] * inv;

    // z1 = relu(h @ fc1_w + fc1_b)
    float z[8];
#pragma unroll
    for (int j = 0; j < 8; ++j) z[j] = s_fc1b[j];
#pragma unroll
    for (int c = 0; c < C; ++c) {
        const float hv = h[c];
#pragma unroll
        for (int j = 0; j < 8; ++j) z[j] = fmaf(hv, s_fc1w[c * 8 + j], z[j]);
    }
#pragma unroll
    for (int j = 0; j < 8; ++j) z[j] = fmaxf(z[j], 0.f);

    // z2 = relu(z1 @ fc2_w + fc2_b); segment max (ReLU>=0 -> int-bit atomicMax valid)
    const int g = batch[node];
#pragma unroll
    for (int k = 0; k < 2; ++k) {
        float acc = s_fc2b[k];
#pragma unroll
        for (int j = 0; j < 8; ++j) acc = fmaf(z[j], s_fc2w[j * 2 + k], acc);
        acc = fmaxf(acc, 0.f);
        atomicMax(&pooled_i[g * 2 + k], __float_as_int(acc));
    }
}

// ---------------- final per-graph softmax over 2 classes ----------------
__global__ void k_final(const int* __restrict__ pooled_i, float* __restrict__ out) {
    int g = threadIdx.x;
    if (g < NGRAPHS) {
        float a = __int_as_float(pooled_i[g * 2 + 0]);
        float b = __int_as_float(pooled_i[g * 2 + 1]);
        float mx = fmaxf(a, b);
        float e0 = expf(a - mx), e1 = expf(b - mx);
        float inv = 1.f / (e0 + e1);
        out[g * 2 + 0] = e0 * inv;
        out[g * 2 + 1] = e1 * inv;
    }
}

extern "C" void kernel_launch(void* const* d_in, const int* in_sizes, int n_in,
                              void* d_out, int out_size, void* d_ws, size_t ws_size,
                              hipStream_t stream) {
    const float* x    = (const float*)d_in[0];
    const int*   eidx = (const int*)d_in[1];
    const int*   bat  = (const int*)d_in[2];
    const float* W1r  = (const float*)d_in[3];
    const float* W1g  = (const float*)d_in[4];
    const float* b1   = (const float*)d_in[5];
    const float* W2r  = (const float*)d_in[6];
    const float* W2g  = (const float*)d_in[7];
    const float* b2   = (const float*)d_in[8];
    const float* fc1w = (const float*)d_in[9];
    const float* fc1b = (const float*)d_in[10];
    const float* fc2w = (const float*)d_in[11];
    const float* fc2b = (const float*)d_in[12];

    const int* src = eidx;             // edge_index[0]
    const int* dst = eidx + NEDGES;    // edge_index[1]

    const size_t featBytes = (size_t)NNODES * C * sizeof(float);  // 12.8 MB
    float* agg    = (float*)d_ws;
    float* h1     = (float*)((char*)d_ws + 1 * featBytes);
    float* h2     = (float*)((char*)d_ws + 2 * featBytes);
    int*   pooled = (int*)  ((char*)d_ws + 3 * featBytes);

    float* out_mid = (float*)d_out;
    float* out_g   = (float*)d_out + (size_t)NNODES * 8;

    const int n4      = NNODES * C / 4;        // 800000 float4s
    const int zblocks = n4 / 256;              // 3125 (exact)
    const int sblocks = (NEDGES * 16) / 256;   // 50000 (exact)
    const int ntiles  = NNODES / 16;           // 3125 (exact, no tail)

    k_zero_i<<<1, 128, 0, stream>>>(pooled, NGRAPHS * 2);
    k_zero_f4<<<zblocks, 256, 0, stream>>>((float4*)agg, n4);
    k_scatter<<<sblocks, 256, 0, stream>>>(x, src, dst, agg);
    k_conv_gemm<<<1024, 128, 0, stream>>>(x, agg, W1r, W1g, b1, h1, ntiles);
    k_zero_f4<<<zblocks, 256, 0, stream>>>((float4*)agg, n4);
    k_scatter<<<sblocks, 256, 0, stream>>>(h1, src, dst, agg);
    k_conv_gemm<<<1024, 128, 0, stream>>>(h1, agg, W2r, W2g, b2, h2, ntiles);
    k_post<<<(NNODES + 255) / 256, 256, 0, stream>>>(h2, bat, fc1w, fc1b, fc2w, fc2b,
                                                     out_mid, pooled);
    k_final<<<1, 64, 0, stream>>>(pooled, out_g);
}